// GGNN_19344532701778
// MI455X (gfx1250) — compile-verified
//
#include <hip/hip_runtime.h>
#include <math.h>

typedef __attribute__((ext_vector_type(2))) float v2f;
typedef __attribute__((ext_vector_type(8))) float v8f;

#define N_NODES 100000
#define N_EDGES 3200000
#define C 50          // channels
#define HS 64         // padded row stride for h/m/agg workspace buffers
#define KP 52         // K padded to multiple of 4
#define KSTEPS 13     // 52 / 4

// fragment-buffer sizes (floats): entries = KSTEPS*NC*2, each entry = v2f
#define FRAG_FLOATS(NC) (KSTEPS * (NC) * 2 * 2)

// D = A(16x4,f32) * B(4x16,f32) + C(16x16,f32), full fp32 WMMA
__device__ __forceinline__ v8f wmma_f32(v2f a, v2f b, v8f c) {
  return __builtin_amdgcn_wmma_f32_16x16x4_f32(
      /*neg_a=*/false, a, /*neg_b=*/false, b,
      /*c_mod=*/(short)0, c, /*reuse_a=*/false, /*reuse_b=*/false);
}

// ---------------------------------------------------------------- utilities
__global__ void k_zero(float* __restrict__ p, int n) {
  int i = blockIdx.x * blockDim.x + threadIdx.x;
  if (i < n) p[i] = 0.0f;
}

// x [N,50] contiguous -> h [N,64] padded (pad cols zeroed)
__global__ void k_stage_x(const float* __restrict__ x, float* __restrict__ h) {
  int i = blockIdx.x * blockDim.x + threadIdx.x;
  if (i < N_NODES * HS) {
    int r = i / HS, c = i % HS;
    h[i] = (c < C) ? x[r * C + c] : 0.0f;
  }
}

// Build fragment-native padded B operand buffer.
// Entry index i = (s*NC + col)*2 + h  ->  v2f { B[4s+2h][col], B[4s+2h+1][col] }
// transposed=0: B[k][col] = W[k*validCols + col]   (conv weight, K x cols)
// transposed=1: B[k][col] = W[col*C + k]           (w @ W.T style weights)
__global__ void k_prep_frag(const float* __restrict__ W, float* __restrict__ F,
                            int NC, int validCols, int transposed) {
  int i = blockIdx.x * blockDim.x + threadIdx.x;
  int total = KSTEPS * NC * 2;
  if (i >= total) return;
  int h   = i & 1;
  int col = (i >> 1) % NC;
  int s   = (i >> 1) / NC;
  int k0  = s * 4 + 2 * h;
  float v0 = 0.0f, v1 = 0.0f;
  if (col < validCols) {
    if (transposed) {
      v0 = (k0     < C) ? W[col * C + k0]     : 0.0f;
      v1 = (k0 + 1 < C) ? W[col * C + k0 + 1] : 0.0f;
    } else {
      v0 = (k0     < C) ? W[k0 * validCols + col]       : 0.0f;
      v1 = (k0 + 1 < C) ? W[(k0 + 1) * validCols + col] : 0.0f;
    }
  }
  F[2 * i]     = v0;
  F[2 * i + 1] = v1;
}

// ------------------------------------------------------- m = h @ W  (50x50)
__global__ __launch_bounds__(32) void k_mm_conv(const float* __restrict__ h,
                                                const v2f* __restrict__ Bf,  // NC=64 frags
                                                float* __restrict__ m) {
  __shared__ float sA[16 * KP];
  const int lane = threadIdx.x;
  const int row0 = blockIdx.x * 16;
  for (int i = lane; i < 16 * KP; i += 32) {
    int r = i / KP, k = i % KP;
    sA[i] = (k < C) ? h[(row0 + r) * HS + k] : 0.0f;
  }
  __syncthreads();
  const int mrow = lane & 15;
  const int half = lane >> 4;
  const v2f* sAv = (const v2f*)sA;          // KP even, pairs 8B aligned
  v2f aF[KSTEPS];
#pragma unroll
  for (int s = 0; s < KSTEPS; ++s)
    aF[s] = sAv[mrow * (KP / 2) + s * 2 + half];
  const int colL = lane & 15;
#pragma unroll
  for (int jt = 0; jt < 4; ++jt) {
    int col = jt * 16 + colL;
    v8f acc = {};
#pragma unroll
    for (int s = 0; s < KSTEPS; ++s)
      acc = wmma_f32(aF[s], Bf[(s * 64 + col) * 2 + half], acc);
#pragma unroll
    for (int v = 0; v < 8; ++v) {
      int r = v + (half << 3);
      m[(row0 + r) * HS + jt * 16 + colL] = acc[v];
    }
  }
}

// ------------------------------------- agg[dst] += m[src]  (wave per edge)
__global__ __launch_bounds__(256) void k_scatter(const float* __restrict__ m,
                                                 const int* __restrict__ ei,
                                                 float* __restrict__ agg) {
  int e = blockIdx.x * 8 + (threadIdx.x >> 5);
  int lane = threadIdx.x & 31;
  if (e >= N_EDGES) return;
  int src = ei[e];
  int dst = ei[N_EDGES + e];
  const float* ms = m + (size_t)src * HS;
  float* ad = agg + (size_t)dst * HS;
  if (lane < C) atomicAdd(&ad[lane], ms[lane]);
  int l2 = lane + 32;
  if (l2 < C) atomicAdd(&ad[l2], ms[l2]);
}

// --------------- fused GRU cell: gi = agg@w_ih^T, gh = h@w_hh^T, gate math
__global__ __launch_bounds__(32) void k_gru(const float* __restrict__ agg,
                                            const float* __restrict__ h,
                                            const v2f* __restrict__ Bi,   // NC=160 frags
                                            const v2f* __restrict__ Bh,   // NC=160 frags
                                            const float* __restrict__ b_ih,
                                            const float* __restrict__ b_hh,
                                            float* __restrict__ hn) {
  __shared__ float sAg[16 * KP];
  __shared__ float sAh[16 * KP];
  __shared__ float sGi[16 * 160];
  __shared__ float sGh[16 * 160];
  const int lane = threadIdx.x;
  const int row0 = blockIdx.x * 16;
  for (int i = lane; i < 16 * KP; i += 32) {
    int r = i / KP, k = i % KP;
    sAg[i] = (k < C) ? agg[(row0 + r) * HS + k] : 0.0f;
    sAh[i] = (k < C) ? h[(row0 + r) * HS + k] : 0.0f;
  }
  __syncthreads();
  const int mrow = lane & 15;
  const int half = lane >> 4;
  const v2f* sAgv = (const v2f*)sAg;
  const v2f* sAhv = (const v2f*)sAh;
  v2f aG[KSTEPS], aH[KSTEPS];
#pragma unroll
  for (int s = 0; s < KSTEPS; ++s) {
    aG[s] = sAgv[mrow * (KP / 2) + s * 2 + half];
    aH[s] = sAhv[mrow * (KP / 2) + s * 2 + half];
  }
  const int colL = lane & 15;
#pragma unroll
  for (int jt = 0; jt < 10; ++jt) {        // 10 * 16 cols covers 3C = 150
    int col = jt * 16 + colL;
    v8f ai = {}, ah = {};
#pragma unroll
    for (int s = 0; s < KSTEPS; ++s) {
      ai = wmma_f32(aG[s], Bi[(s * 160 + col) * 2 + half], ai);
      ah = wmma_f32(aH[s], Bh[(s * 160 + col) * 2 + half], ah);
    }
#pragma unroll
    for (int v = 0; v < 8; ++v) {
      int r = v + (half << 3);
      sGi[r * 160 + jt * 16 + colL] = ai[v];
      sGh[r * 160 + jt * 16 + colL] = ah[v];
    }
  }
  __syncthreads();
  // gates: 16 rows x 50 cols = 800 elems, 25 per lane, no divergence
  for (int idx = lane; idx < 16 * C; idx += 32) {
    int r = idx / C, c = idx % C;
    float i_r = sGi[r * 160 + c]         + b_ih[c];
    float i_z = sGi[r * 160 + c + C]     + b_ih[c + C];
    float i_n = sGi[r * 160 + c + 2 * C] + b_ih[c + 2 * C];
    float h_r = sGh[r * 160 + c]         + b_hh[c];
    float h_z = sGh[r * 160 + c + C]     + b_hh[c + C];
    float h_n = sGh[r * 160 + c + 2 * C] + b_hh[c + 2 * C];
    float rr = 1.0f / (1.0f + __expf(-(i_r + h_r)));
    float zz = 1.0f / (1.0f + __expf(-(i_z + h_z)));
    float nn = tanhf(i_n + rr * h_n);
    float hv = h[(row0 + r) * HS + c];
    hn[(row0 + r) * HS + c] = (1.0f - zz) * nn + zz * hv;
  }
}

// ---------------------------- out = relu(h) @ lin_w^T + lin_b  (stride-50)
__global__ __launch_bounds__(32) void k_out(const float* __restrict__ h,
                                            const v2f* __restrict__ Bf,  // NC=64 frags
                                            const float* __restrict__ lin_b,
                                            float* __restrict__ out) {
  __shared__ float sA[16 * KP];
  const int lane = threadIdx.x;
  const int row0 = blockIdx.x * 16;
  for (int i = lane; i < 16 * KP; i += 32) {
    int r = i / KP, k = i % KP;
    float v = (k < C) ? h[(row0 + r) * HS + k] : 0.0f;
    sA[i] = fmaxf(v, 0.0f);                // fused ReLU
  }
  __syncthreads();
  const int mrow = lane & 15;
  const int half = lane >> 4;
  const v2f* sAv = (const v2f*)sA;
  v2f aF[KSTEPS];
#pragma unroll
  for (int s = 0; s < KSTEPS; ++s)
    aF[s] = sAv[mrow * (KP / 2) + s * 2 + half];
  const int colL = lane & 15;
#pragma unroll
  for (int jt = 0; jt < 4; ++jt) {
    int col = jt * 16 + colL;
    bool cv = col < C;
    v8f acc = {};
#pragma unroll
    for (int s = 0; s < KSTEPS; ++s)
      acc = wmma_f32(aF[s], Bf[(s * 64 + col) * 2 + half], acc);
    float bias = cv ? lin_b[col] : 0.0f;
#pragma unroll
    for (int v = 0; v < 8; ++v) {
      int r = v + (half << 3);
      if (cv) out[(size_t)(row0 + r) * C + col] = acc[v] + bias;
    }
  }
}

// ----------------------------------------------------------------- launcher
extern "C" void kernel_launch(void* const* d_in, const int* in_sizes, int n_in,
                              void* d_out, int out_size, void* d_ws, size_t ws_size,
                              hipStream_t stream) {
  const float* x     = (const float*)d_in[0];
  const int*   ei    = (const int*)d_in[1];
  // d_in[2] edge_attr, d_in[8] edge_emb: unused in reference forward
  const float* convW = (const float*)d_in[3];
  const float* w_ih  = (const float*)d_in[4];
  const float* w_hh  = (const float*)d_in[5];
  const float* b_ih  = (const float*)d_in[6];
  const float* b_hh  = (const float*)d_in[7];
  const float* lin_w = (const float*)d_in[9];
  const float* lin_b = (const float*)d_in[10];
  float* out = (float*)d_out;

  const size_t nbuf = (size_t)N_NODES * HS;
  float* bufA  = (float*)d_ws;
  float* bufB  = bufA + nbuf;
  float* agg   = bufB + nbuf;
  float* fConv = agg + nbuf;                       // 3 layers, NC=64
  float* fIh   = fConv + 3 * FRAG_FLOATS(64);      // NC=160
  float* fHh   = fIh + FRAG_FLOATS(160);           // NC=160
  float* fLin  = fHh + FRAG_FLOATS(160);           // NC=64

  // ---- prep: fragment-native padded weight buffers (tiny, L2-resident)
  {
    int t64  = KSTEPS * 64 * 2;    // entries for NC=64
    int t160 = KSTEPS * 160 * 2;   // entries for NC=160
    for (int l = 0; l < 3; ++l)
      k_prep_frag<<<(t64 + 255) / 256, 256, 0, stream>>>(
          convW + (size_t)l * C * C, fConv + (size_t)l * FRAG_FLOATS(64), 64, C, 0);
    k_prep_frag<<<(t160 + 255) / 256, 256, 0, stream>>>(w_ih, fIh, 160, 3 * C, 1);
    k_prep_frag<<<(t160 + 255) / 256, 256, 0, stream>>>(w_hh, fHh, 160, 3 * C, 1);
    k_prep_frag<<<(t64 + 255) / 256, 256, 0, stream>>>(lin_w, fLin, 64, C, 1);
  }

  float* h = bufA;
  float* m = bufB;
  const int tiles  = N_NODES / 16;     // 6250, exact
  const int nElems = N_NODES * HS;

  k_stage_x<<<(nElems + 255) / 256, 256, 0, stream>>>(x, h);

  for (int layer = 0; layer < 3; ++layer) {
    k_mm_conv<<<tiles, 32, 0, stream>>>(
        h, (const v2f*)(fConv + (size_t)layer * FRAG_FLOATS(64)), m);
    k_zero<<<(nElems + 255) / 256, 256, 0, stream>>>(agg, nElems);
    k_scatter<<<N_EDGES / 8, 256, 0, stream>>>(m, ei, agg);
    k_gru<<<tiles, 32, 0, stream>>>(agg, h, (const v2f*)fIh, (const v2f*)fHh,
                                    b_ih, b_hh, m);
    float* t = h; h = m; m = t;        // ping-pong h / h_next
  }
  k_out<<<tiles, 32, 0, stream>>>(h, (const v2f*)fLin, lin_b, out);
}